// stackedIndRNN_encoder_1228360646807
// MI455X (gfx1250) — compile-verified
//
#include <hip/hip_runtime.h>
#include <hip/hip_bf16.h>

typedef __attribute__((ext_vector_type(2))) float v2f;
typedef __attribute__((ext_vector_type(8))) float v8f;

#define EPSV 1e-5f

// ---------------- embedding gather ----------------
__global__ void embed_kernel(const int* __restrict__ tokens,
                             const float* __restrict__ emb,
                             float* __restrict__ X, int SB, int E) {
  int i = blockIdx.x * blockDim.x + threadIdx.x;
  if (i >= SB * E) return;
  int r = i / E;
  int e = i - r * E;
  X[i] = emb[(size_t)tokens[r] * (size_t)E + e];
}

// ---------------- fp32 WMMA GEMM: C[M,N] = A[M,K] @ B[K,N] + bias ----------------
// block = 256 threads (8 waves). Block tile 256(M) x 64(N).
// Each wave computes a 32(M) x 64(N) strip: 2x4 accumulators of 16x16.
// Requires M % 256 == 0, K % 16 == 0, N % 16 == 0 (true for all GEMMs here).
#define BM 256
#define BN 64
#define BK 16
#define AP 20   // padded A row stride (bank-conflict-free fragment reads)
#define BNP 80  // padded B pair-row stride (x2 floats per col)

__global__ __launch_bounds__(256)
void gemm_bias_wmma(const float* __restrict__ A, const float* __restrict__ Bmat,
                    const float* __restrict__ bias, float* __restrict__ C,
                    int M, int K, int N) {
  __shared__ float As[BM * AP];               // 256 x 16, padded rows
  __shared__ float Bs[(BK / 2) * BNP * 2];    // pair-interleaved: ((r>>1)*BNP + c)*2 + (r&1)

  const int tid  = threadIdx.x;
  const int wave = tid >> 5;
  const int lane = tid & 31;
  const int half = lane >> 4;     // 0: lanes 0-15, 1: lanes 16-31
  const int lr   = lane & 15;
  const int kof  = half * 2;      // K offset per ISA A/B fragment layout

  const int mBlock = blockIdx.y * BM;
  const int nBlock = blockIdx.x * BN;

  // staging indices: A tile 256x16 as 4 float4/thread; B tile 16x64 as 1 float4/thread
  const int ar = tid >> 2;            // A row (plus i*64)
  const int ac = (tid & 3) * 4;       // A col quad
  const int brow = tid >> 4;          // B row
  const int bc = (tid & 15) * 4;      // B col quad
  const bool bColOk = (nBlock + bc) < N;  // N % 16 == 0 -> whole quad in range

  // global source pointers for this thread's staging quads
  const float* aSrc0 = A + (size_t)(mBlock + ar +   0) * (size_t)K + ac;
  const float* aSrc1 = A + (size_t)(mBlock + ar +  64) * (size_t)K + ac;
  const float* aSrc2 = A + (size_t)(mBlock + ar + 128) * (size_t)K + ac;
  const float* aSrc3 = A + (size_t)(mBlock + ar + 192) * (size_t)K + ac;
  const float* bSrc  = Bmat + (size_t)brow * (size_t)N + (nBlock + bc);

  // LDS destinations
  float* aDst0 = &As[(ar +   0) * AP + ac];
  float* aDst1 = &As[(ar +  64) * AP + ac];
  float* aDst2 = &As[(ar + 128) * AP + ac];
  float* aDst3 = &As[(ar + 192) * AP + ac];
  const int pr = brow >> 1;
  const int wi = brow & 1;
  float* bDst = &Bs[(pr * BNP + bc) * 2 + wi];

  v8f acc[2][4] = {};

  // named staging registers (no arrays/lambdas -> stay in VGPRs)
  float4 aR0, aR1, aR2, aR3, bR;

  // prologue: prefetch first K-tile
  aR0 = *reinterpret_cast<const float4*>(aSrc0);
  aR1 = *reinterpret_cast<const float4*>(aSrc1);
  aR2 = *reinterpret_cast<const float4*>(aSrc2);
  aR3 = *reinterpret_cast<const float4*>(aSrc3);
  bR  = bColOk ? *reinterpret_cast<const float4*>(bSrc)
               : make_float4(0.0f, 0.0f, 0.0f, 0.0f);

  for (int kt = 0; kt < K; kt += BK) {
    // stage current tile into LDS
    *reinterpret_cast<float4*>(aDst0) = aR0;
    *reinterpret_cast<float4*>(aDst1) = aR1;
    *reinterpret_cast<float4*>(aDst2) = aR2;
    *reinterpret_cast<float4*>(aDst3) = aR3;
    bDst[0] = bR.x;
    bDst[2] = bR.y;
    bDst[4] = bR.z;
    bDst[6] = bR.w;
    __syncthreads();

    // prefetch next tile behind this tile's compute
    if (kt + BK < K) {
      const int ko = kt + BK;
      aR0 = *reinterpret_cast<const float4*>(aSrc0 + ko);
      aR1 = *reinterpret_cast<const float4*>(aSrc1 + ko);
      aR2 = *reinterpret_cast<const float4*>(aSrc2 + ko);
      aR3 = *reinterpret_cast<const float4*>(aSrc3 + ko);
      bR  = bColOk ? *reinterpret_cast<const float4*>(bSrc + (size_t)ko * (size_t)N)
                   : make_float4(0.0f, 0.0f, 0.0f, 0.0f);
    }

    const int aRow0 = wave * 32 + lr;
#pragma unroll
    for (int kk = 0; kk < BK; kk += 4) {
      // A fragments (16x4): lanes 0-15 -> K={kk,kk+1}; lanes 16-31 -> K={kk+2,kk+3}
      v2f a0 = *reinterpret_cast<const v2f*>(&As[aRow0 * AP + kk + kof]);
      v2f a1 = *reinterpret_cast<const v2f*>(&As[(aRow0 + 16) * AP + kk + kof]);
#pragma unroll
      for (int j = 0; j < 4; ++j) {
        // B fragment (4x16): one aligned b64 from pair-interleaved layout
        v2f b = *reinterpret_cast<const v2f*>(
            &Bs[(((kk >> 1) + half) * BNP + j * 16 + lr) * 2]);
        acc[0][j] = __builtin_amdgcn_wmma_f32_16x16x4_f32(
            false, a0, false, b, (short)0, acc[0][j], false, false);
        acc[1][j] = __builtin_amdgcn_wmma_f32_16x16x4_f32(
            false, a1, false, b, (short)0, acc[1][j], false, false);
      }
    }
    __syncthreads();
  }

  // writeout: VGPR r holds (M=r, lanes 0-15) and (M=r+8, lanes 16-31)
#pragma unroll
  for (int mi = 0; mi < 2; ++mi) {
    const int rowBase = mBlock + wave * 32 + mi * 16 + half * 8;
#pragma unroll
    for (int j = 0; j < 4; ++j) {
      int col = nBlock + j * 16 + lr;
      if (col < N) {
        float bs = bias[col];
#pragma unroll
        for (int r = 0; r < 8; ++r) {
          C[(size_t)(rowBase + r) * (size_t)N + col] = acc[mi][j][r] + bs;
        }
      }
    }
  }
}

// ---------------- fused eval-BN + IndRNN scan over S ----------------
// Y,X: [S, rowElems] where rowElems = B*F ; one thread per (b,f)
__global__ void bn_indrnn_scan(const float* __restrict__ Y, float* __restrict__ X,
                               const float* __restrict__ g, const float* __restrict__ be,
                               const float* __restrict__ mu, const float* __restrict__ var,
                               const float* __restrict__ u,
                               int S, int rowElems, int F) {
  int t = blockIdx.x * blockDim.x + threadIdx.x;
  if (t >= rowElems) return;
  int f = t % F;
  float sc = g[f] * rsqrtf(var[f] + EPSV);
  float sh = fmaf(-mu[f], sc, be[f]);
  float uu = u[f];
  float h = 0.0f;
  const float* y = Y + t;
  float* x = X + t;
  for (int s = 0; s < S; ++s) {
    float v = fmaf(*y, sc, sh);      // batch-norm affine
    h = fmaxf(fmaf(uu, h, v), 0.0f); // h = relu(v + u*h)
    *x = h;
    y += rowElems;
    x += rowElems;
  }
}

// ---------------- plain eval-BN (extra_bn) ----------------
__global__ void bn_kernel(const float* __restrict__ Xin, float* __restrict__ Xout,
                          const float* __restrict__ g, const float* __restrict__ be,
                          const float* __restrict__ mu, const float* __restrict__ var,
                          int total, int F) {
  int i = blockIdx.x * blockDim.x + threadIdx.x;
  if (i >= total) return;
  int f = i % F;
  float sc = g[f] * rsqrtf(var[f] + EPSV);
  Xout[i] = fmaf(Xin[i] - mu[f], sc, be[f]);
}

extern "C" void kernel_launch(void* const* d_in, const int* in_sizes, int n_in,
                              void* d_out, int out_size, void* d_ws, size_t ws_size,
                              hipStream_t stream) {
  (void)in_sizes; (void)n_in; (void)out_size; (void)ws_size;
  const int S = 256, B = 32, H = 2000, E = 512, V = 10000, L = 6;
  const int SB = S * B;  // 8192 rows

  const int*   tokens = (const int*)  d_in[0];
  const float* emb    = (const float*)d_in[1];
  const float* W0     = (const float*)d_in[2];
  const float* b0     = (const float*)d_in[3];
  const float* Wr     = (const float*)d_in[4];   // [L-1, H, H]
  const float* br     = (const float*)d_in[5];   // [L-1, H]
  const float* u      = (const float*)d_in[6];   // [L, H]
  const float* g      = (const float*)d_in[7];
  const float* be     = (const float*)d_in[8];
  const float* mu     = (const float*)d_in[9];
  const float* var    = (const float*)d_in[10];
  const float* Wl     = (const float*)d_in[11];  // [H, E]
  const float* bl     = (const float*)d_in[12];
  const float* ul     = (const float*)d_in[13];
  const float* gl     = (const float*)d_in[14];
  const float* bel    = (const float*)d_in[15];
  const float* mul    = (const float*)d_in[16];
  const float* varl   = (const float*)d_in[17];
  const float* g2     = (const float*)d_in[18];
  const float* be2    = (const float*)d_in[19];
  const float* mu2    = (const float*)d_in[20];
  const float* var2   = (const float*)d_in[21];
  const float* Wd     = (const float*)d_in[22];  // [E, V]
  const float* bd     = (const float*)d_in[23];

  float* out  = (float*)d_out;
  float* bufA = (float*)d_ws;                    // [SB, H] max
  float* bufB = bufA + (size_t)SB * (size_t)H;   // [SB, H] max

  const int THREADS = 256;

  // 1) embedding gather -> bufA [SB, E]
  {
    int total = SB * E;
    embed_kernel<<<(total + THREADS - 1) / THREADS, THREADS, 0, stream>>>(
        tokens, emb, bufA, SB, E);
  }

  // 2) layer 0: GEMM [SB,E]@[E,H] -> bufB, then BN+scan -> bufA
  {
    dim3 grid((H + BN - 1) / BN, SB / BM);
    gemm_bias_wmma<<<grid, THREADS, 0, stream>>>(bufA, W0, b0, bufB, SB, E, H);
    int rowElems = B * H;
    bn_indrnn_scan<<<(rowElems + THREADS - 1) / THREADS, THREADS, 0, stream>>>(
        bufB, bufA, g, be, mu, var, u, S, rowElems, H);
  }

  // 3) layers 1..5: GEMM [SB,H]@[H,H] -> bufB, BN+scan -> bufA
  for (int l = 1; l < L; ++l) {
    dim3 grid((H + BN - 1) / BN, SB / BM);
    const float* Wrl = Wr + (size_t)(l - 1) * (size_t)H * (size_t)H;
    const float* brl = br + (size_t)(l - 1) * (size_t)H;
    gemm_bias_wmma<<<grid, THREADS, 0, stream>>>(bufA, Wrl, brl, bufB, SB, H, H);
    int rowElems = B * H;
    bn_indrnn_scan<<<(rowElems + THREADS - 1) / THREADS, THREADS, 0, stream>>>(
        bufB, bufA, g + (size_t)l * H, be + (size_t)l * H,
        mu + (size_t)l * H, var + (size_t)l * H, u + (size_t)l * H,
        S, rowElems, H);
  }

  // 4) last_fc: [SB,H]@[H,E] -> bufB, BN+scan -> bufA [SB,E]
  {
    dim3 grid((E + BN - 1) / BN, SB / BM);
    gemm_bias_wmma<<<grid, THREADS, 0, stream>>>(bufA, Wl, bl, bufB, SB, H, E);
    int rowElems = B * E;
    bn_indrnn_scan<<<(rowElems + THREADS - 1) / THREADS, THREADS, 0, stream>>>(
        bufB, bufA, gl, bel, mul, varl, ul, S, rowElems, E);
  }

  // 5) extra_bn: bufA -> bufB [SB,E]
  {
    int total = SB * E;
    bn_kernel<<<(total + THREADS - 1) / THREADS, THREADS, 0, stream>>>(
        bufA, bufB, g2, be2, mu2, var2, total, E);
  }

  // 6) decoder: [SB,E]@[E,V] + bd -> out [SB,V]
  {
    dim3 grid((V + BN - 1) / BN, SB / BM);
    gemm_bias_wmma<<<grid, THREADS, 0, stream>>>(bufB, Wd, bd, out, SB, E, V);
  }
}